// RackioObserverLSTM_f_Q_R_H_87041807221158
// MI455X (gfx1250) — compile-verified
//
#include <hip/hip_runtime.h>
#include <cmath>

// Problem constants (match reference)
#define BB 256   // batch
#define TT 512   // timesteps
#define FF 128   // input features
#define UU 512   // LSTM units
#define G4 2048  // 4*UU (gates i,f,g,o)
#define KK 640   // UU + FF  (A = [h | x_t], WcT = [U ; W]^T)

typedef __attribute__((ext_vector_type(2))) float v2f;
typedef __attribute__((ext_vector_type(8))) float v8f;

// workspace layout (in floats)
#define WS_WCT 0                      // WcT: [G4][KK] = 2048 x 640
#define WS_H0  (G4 * KK)              // 1,310,720
#define WS_H1  (WS_H0 + BB * UU)
#define WS_C   (WS_H1 + BB * UU)      // total ~6.8 MB

// ---------------------------------------------------------------- prep ------
// Build transposed stacked weights: WcT[n][k] = (k<UU ? U[k][n] : W[k-UU][n]).
// One-time cost; makes K the contiguous dim so B fragments are single b64
// LDS loads in the hot loop (same shape as A fragments).
__global__ void build_wct(const float* __restrict__ W, const float* __restrict__ U,
                          float* __restrict__ WcT) {
    int idx = blockIdx.x * blockDim.x + threadIdx.x;
    if (idx >= KK * G4) return;
    int n = idx / KK, k = idx % KK;
    WcT[idx] = (k < UU) ? U[k * G4 + n] : W[(k - UU) * G4 + n];
}

__global__ void zero_state(float* __restrict__ h0, float* __restrict__ c) {
    int idx = blockIdx.x * blockDim.x + threadIdx.x;
    if (idx < BB * UU) { h0[idx] = 0.0f; c[idx] = 0.0f; }
}

// ------------------------------------------------------------- LSTM step ----
// One workgroup: 32 batch rows x 32 unit cols, all 4 gates (one per wave).
// z = [h|x_t] @ [U;W] + b  via V_WMMA_F32_16X16X4_F32, K staged through LDS
// in chunks of 32.
__launch_bounds__(128)
__global__ void lstm_step(const float* __restrict__ x, const float* __restrict__ WcT,
                          const float* __restrict__ bvec,
                          const float* __restrict__ hprev, float* __restrict__ hnext,
                          float* __restrict__ cbuf, int t) {
    __shared__ float As[32 * 36];          // A tile: [m=32][k=32], stride 36 (pad)
    __shared__ float Bs[4 * 32 * 36];      // B tiles per gate: [n=32][k=32]
    __shared__ float Zs[4 * 32 * 36];      // gate results for cross-wave gating

    const int tid  = threadIdx.x;
    const int lane = tid & 31;
    const int wave = tid >> 5;             // wave == gate id (i,f,g,o)
    const int hlf  = lane >> 4;            // 0 or 1 (wave32 half)
    const int ln16 = lane & 15;
    const int m0   = blockIdx.y * 32;      // batch-row block
    const int u0   = blockIdx.x * 32;      // unit-col block

    v8f acc[2][2] = {};                    // 2x2 tiles of 16x16 fp32

    for (int kc = 0; kc < KK / 32; ++kc) {
        const int kbase = kc * 32;
        // ---- stage A chunk (rows m0..m0+31, k kbase..kbase+31) ----
        #pragma unroll
        for (int it = 0; it < 2; ++it) {
            int task = tid + it * 128;                 // 256 float4 tasks
            int row  = task >> 3;
            int qc   = (task & 7) * 4;
            float4 v;
            if (kbase < UU) {
                v = *(const float4*)(hprev + (size_t)(m0 + row) * UU + kbase + qc);
            } else {
                v = *(const float4*)(x + (size_t)(m0 + row) * (TT * FF)
                                       + (size_t)t * FF + (kbase - UU) + qc);
            }
            *(float4*)(As + row * 36 + qc) = v;
        }
        // ---- stage B chunk for all 4 gates, [n][k] layout ----
        #pragma unroll
        for (int it = 0; it < 8; ++it) {
            int task = tid + it * 128;                 // 1024 float4 tasks
            int g    = task >> 8;
            int rem  = task & 255;
            int nrow = rem >> 3;                       // unit-column within tile
            int qk   = (rem & 7) * 4;                  // k quad within chunk
            float4 v = *(const float4*)(WcT + (size_t)(g * UU + u0 + nrow) * KK
                                            + kbase + qk);
            *(float4*)(Bs + (g * 32 + nrow) * 36 + qk) = v;
        }
        __syncthreads();

        // ---- accumulate: 8 x (K=4) WMMA slices per chunk ----
        const float* BgT = Bs + wave * 32 * 36;
        #pragma unroll
        for (int k4 = 0; k4 < 32; k4 += 4) {
            const int kk = k4 + 2 * hlf;
            // A frags (ISA: lane=M, VGPR0/1 = K = 2*half + {0,1})
            v2f a0 = *(const v2f*)(As + ln16 * 36 + kk);
            v2f a1 = *(const v2f*)(As + (16 + ln16) * 36 + kk);
            // B frags (lane=N, VGPR0/1 = K = 2*half + {0,1}); contiguous in k
            v2f b0 = *(const v2f*)(BgT + ln16 * 36 + kk);
            v2f b1 = *(const v2f*)(BgT + (16 + ln16) * 36 + kk);
            acc[0][0] = __builtin_amdgcn_wmma_f32_16x16x4_f32(false, a0, false, b0,
                            (short)0, acc[0][0], false, false);
            acc[0][1] = __builtin_amdgcn_wmma_f32_16x16x4_f32(false, a0, false, b1,
                            (short)0, acc[0][1], false, false);
            acc[1][0] = __builtin_amdgcn_wmma_f32_16x16x4_f32(false, a1, false, b0,
                            (short)0, acc[1][0], false, false);
            acc[1][1] = __builtin_amdgcn_wmma_f32_16x16x4_f32(false, a1, false, b1,
                            (short)0, acc[1][1], false, false);
        }
        __syncthreads();
    }

    // ---- spill gate tiles to LDS (C/D layout: VGPR v -> M = v + 8*half) ----
    float* Zg = Zs + wave * 32 * 36;
    #pragma unroll
    for (int mi = 0; mi < 2; ++mi)
        #pragma unroll
        for (int ni = 0; ni < 2; ++ni)
            #pragma unroll
            for (int v = 0; v < 8; ++v) {
                int row = mi * 16 + hlf * 8 + v;
                int col = ni * 16 + ln16;
                Zg[row * 36 + col] = acc[mi][ni][v];
            }
    __syncthreads();

    // ---- gating: c = sig(f)*c + sig(i)*g ; h = sig(o)*c (linear cell) ----
    #pragma unroll
    for (int j = 0; j < 8; ++j) {
        int e = tid + j * 128;                 // 1024 elements of the 32x32 tile
        int m = e >> 5;
        int u = e & 31;
        float zi = Zs[0 * 32 * 36 + m * 36 + u] + bvec[0 * UU + u0 + u];
        float zf = Zs[1 * 32 * 36 + m * 36 + u] + bvec[1 * UU + u0 + u];
        float zg = Zs[2 * 32 * 36 + m * 36 + u] + bvec[2 * UU + u0 + u];
        float zo = Zs[3 * 32 * 36 + m * 36 + u] + bvec[3 * UU + u0 + u];
        float si = 1.0f / (1.0f + __expf(-zi));
        float sf = 1.0f / (1.0f + __expf(-zf));
        float so = 1.0f / (1.0f + __expf(-zo));
        int gi = (m0 + m) * UU + u0 + u;
        float cn = sf * cbuf[gi] + si * zg;
        cbuf[gi]  = cn;
        hnext[gi] = so * cn;
    }
}

// ------------------------------------------------------------- epilogue -----
__global__ void dense_out(const float* __restrict__ h, const float* __restrict__ Wd,
                          const float* __restrict__ bd, float* __restrict__ out) {
    int idx = blockIdx.x * blockDim.x + threadIdx.x;
    if (idx >= BB * 7) return;
    int b = idx / 7, j = idx % 7;
    float s = bd[j];
    for (int u = 0; u < UU; ++u)
        s += tanhf(h[b * UU + u]) * Wd[u * 7 + j];
    out[idx] = s;
}

// --------------------------------------------------------------- launch -----
extern "C" void kernel_launch(void* const* d_in, const int* in_sizes, int n_in,
                              void* d_out, int out_size, void* d_ws, size_t ws_size,
                              hipStream_t stream) {
    const float* x  = (const float*)d_in[0];   // [B,T,F]
    const float* W  = (const float*)d_in[1];   // [F,4U]
    const float* U  = (const float*)d_in[2];   // [U,4U]
    const float* b  = (const float*)d_in[3];   // [4U]
    const float* Wd = (const float*)d_in[4];   // [U,7]
    const float* bd = (const float*)d_in[5];   // [7]
    float* out = (float*)d_out;

    float* ws  = (float*)d_ws;
    float* WcT = ws + WS_WCT;
    float* h0  = ws + WS_H0;
    float* h1  = ws + WS_H1;
    float* cb  = ws + WS_C;

    build_wct<<<(KK * G4 + 255) / 256, 256, 0, stream>>>(W, U, WcT);
    zero_state<<<(BB * UU + 255) / 256, 256, 0, stream>>>(h0, cb);

    dim3 grid(UU / 32, BB / 32);               // 16 x 8 = 128 workgroups
    for (int t = 0; t < TT; ++t) {
        const float* hp = (t & 1) ? h1 : h0;
        float*       hn = (t & 1) ? h0 : h1;
        lstm_step<<<grid, 128, 0, stream>>>(x, WcT, b, hp, hn, cb, t);
    }
    // after 512 steps (t=511 writes h0), final hidden state is in h0
    dense_out<<<(BB * 7 + 127) / 128, 128, 0, stream>>>(h0, Wd, bd, out);
}